// ChebyshevConvolutionLin_72911364817014
// MI455X (gfx1250) — compile-verified
//
#include <hip/hip_runtime.h>
#include <math.h>

// ---------------------------------------------------------------------------
// ChebConv(64->64,K=3) -> ReLU -> ChebConv(64->64,K=3) -> Linear(64->16)
//   -> log_softmax ; plus pass-through of edge_index.
// MI455X / gfx1250: wave32, WMMA f32 16x16x4 for the dense GEMMs (fp32 data
// -> fp32 matrix path), hardware f32 global atomics for the SpMM scatter.
// Weights staged pre-swizzled in LDS (B-fragment layout) -> 1 ds_load_b64
// per WMMA B operand instead of 2 strided global_load_b32.
// ---------------------------------------------------------------------------

typedef __attribute__((ext_vector_type(2))) float v2f;
typedef __attribute__((ext_vector_type(8))) float v8f;

#define F_DIM 64
#define CLS   16
#define KCHEB 3

// ---------------- utility: zero a float buffer (float4 wide) ----------------
__global__ void zero_kernel(float4* __restrict__ p, long n4) {
    long i = (long)blockIdx.x * blockDim.x + threadIdx.x;
    if (i < n4) p[i] = make_float4(0.f, 0.f, 0.f, 0.f);
}

// ---------------- degree: deg[src] += 1 for non-self-loop edges -------------
__global__ void degree_kernel(const int* __restrict__ src,
                              const int* __restrict__ dst,
                              float* __restrict__ deg, int E) {
    int e = blockIdx.x * blockDim.x + threadIdx.x;
    if (e >= E) return;
    int s = src[e];
    if (s != dst[e]) unsafeAtomicAdd(&deg[s], 1.0f);
}

// ---------------- dinv[i] = deg>0 ? 1/sqrt(deg) : 0 (in place) --------------
__global__ void dinv_kernel(float* __restrict__ d, int n) {
    int i = blockIdx.x * blockDim.x + threadIdx.x;
    if (i >= n) return;
    float v = d[i];
    d[i] = (v > 0.f) ? rsqrtf(v) : 0.f;
}

// ---------------- norm[e] = -dinv[src]*dinv[dst] (0 on self loop) -----------
__global__ void norm_kernel(const int* __restrict__ src,
                            const int* __restrict__ dst,
                            const float* __restrict__ dinv,
                            float* __restrict__ norm, int E) {
    int e = blockIdx.x * blockDim.x + threadIdx.x;
    if (e >= E) return;
    int s = src[e], d = dst[e];
    norm[e] = (s == d) ? 0.f : -dinv[s] * dinv[d];
}

// ---------------- SpMM: out[dst] += norm * in[src], one wave per edge -------
// 32 lanes x float2 = 64 features per edge. Dense node matrices (25.6MB)
// stay resident in the 192MB L2, so gather + f32-atomic scatter is on-chip.
__global__ __launch_bounds__(256) void spmm_kernel(
    const int* __restrict__ src, const int* __restrict__ dst,
    const float* __restrict__ norm, const float* __restrict__ xin,
    float* __restrict__ xout, int E) {
    int wid  = (int)((blockIdx.x * (long)blockDim.x + threadIdx.x) >> 5);
    int lane = threadIdx.x & 31;
    if (wid >= E) return;
    float nm = norm[wid];
    if (nm == 0.f) return;
    int s = src[wid], d = dst[wid];
    const float2* xi = (const float2*)(xin + (size_t)s * F_DIM);
    float2 v = xi[lane];
    float* o = xout + (size_t)d * F_DIM + lane * 2;
    unsafeAtomicAdd(o,     nm * v.x);
    unsafeAtomicAdd(o + 1, nm * v.y);
}

// ---------------- Chebyshev recurrence combine: p = 2*p - t0 ----------------
__global__ void combine_kernel(float4* __restrict__ p,
                               const float4* __restrict__ t0, long n4) {
    long i = (long)blockIdx.x * blockDim.x + threadIdx.x;
    if (i < n4) {
        float4 a = p[i], b = t0[i];
        p[i] = make_float4(2.f * a.x - b.x, 2.f * a.y - b.y,
                           2.f * a.z - b.z, 2.f * a.w - b.w);
    }
}

// ---------------- fused ChebConv GEMM:  out = sum_k tx_k @ W[k] + b ---------
// One wave owns a 16(M) x 64(N) output strip: 4 f32 accumulators (8 VGPRs
// each, bias splatted in), A fragment reused across the 4 N-tiles,
// 3 orders x 16 k-steps x 4 n-tiles = 192 v_wmma_f32_16x16x4_f32 per wave.
// Weights pre-swizzled into LDS in B-fragment layout: element [kk][ks][j][L]
// = (W[kk][ks*4 + 2*(L>=16)][j*16 + L%16], W[kk][ks*4 + 2*(L>=16)+1][...]).
// VGPR layouts per CDNA5 ISA 7.12.2. Requires N % 16 == 0 (100000 = 6250*16).
__global__ __launch_bounds__(256) void cheb_gemm_kernel(
    const float* __restrict__ tx0, const float* __restrict__ tx1,
    const float* __restrict__ tx2,
    const float* __restrict__ W,      // [3][64][64] (k, f_in, f_out)
    const float* __restrict__ bias,   // [64]
    float* __restrict__ out, int mtiles, int do_relu) {
    __shared__ float2 sw[KCHEB * 16 * 4 * 32];   // 48 KB (WGP has 320 KB)

    // cooperative pre-swizzle of weights into the WMMA B-fragment layout
    for (int m = threadIdx.x; m < KCHEB * 16 * 4 * 32; m += blockDim.x) {
        int L   = m & 31;
        int j   = (m >> 5) & 3;
        int ks  = (m >> 7) & 15;
        int kk  = m >> 11;
        int k0  = ks * 4 + ((L >> 4) << 1);
        int col = j * 16 + (L & 15);
        const float* Wk = W + kk * F_DIM * F_DIM + (size_t)k0 * F_DIM + col;
        sw[m] = make_float2(Wk[0], Wk[F_DIM]);
    }
    __syncthreads();

    int wave  = threadIdx.x >> 5;
    int lane  = threadIdx.x & 31;
    int mtile = blockIdx.x * 8 + wave;
    if (mtile >= mtiles) return;     // after the barrier: no divergence hazard

    int half = lane >> 4;
    int l16  = lane & 15;
    int arow = mtile * 16 + l16;     // A-matrix row handled by this lane

    // bias folded into accumulator init: every C element of n-tile j at this
    // lane is column j*16+l16, so the splat applies bias exactly once.
    float b0 = bias[l16], b1 = bias[16 + l16], b2 = bias[32 + l16], b3 = bias[48 + l16];
    v8f acc0 = {b0, b0, b0, b0, b0, b0, b0, b0};
    v8f acc1 = {b1, b1, b1, b1, b1, b1, b1, b1};
    v8f acc2 = {b2, b2, b2, b2, b2, b2, b2, b2};
    v8f acc3 = {b3, b3, b3, b3, b3, b3, b3, b3};

    const float* txs[KCHEB] = {tx0, tx1, tx2};
#pragma unroll
    for (int kk = 0; kk < KCHEB; ++kk) {
        const float* A = txs[kk] + (size_t)arow * F_DIM;
        const float2* swk = sw + (kk << 11) + lane;   // + ks*128 + j*32
#pragma unroll
        for (int ks = 0; ks < F_DIM / 4; ++ks) {
            int k0 = ks * 4 + half * 2;
            v2f a;
            a.x = A[k0];
            a.y = A[k0 + 1];
            const float2* sb = swk + (ks << 7);
            float2 f0 = sb[0];
            float2 f1 = sb[32];
            float2 f2 = sb[64];
            float2 f3 = sb[96];
            v2f b;
            b.x = f0.x; b.y = f0.y;
            acc0 = __builtin_amdgcn_wmma_f32_16x16x4_f32(false, a, false, b, (short)0, acc0, false, false);
            b.x = f1.x; b.y = f1.y;
            acc1 = __builtin_amdgcn_wmma_f32_16x16x4_f32(false, a, false, b, (short)0, acc1, false, false);
            b.x = f2.x; b.y = f2.y;
            acc2 = __builtin_amdgcn_wmma_f32_16x16x4_f32(false, a, false, b, (short)0, acc2, false, false);
            b.x = f3.x; b.y = f3.y;
            acc3 = __builtin_amdgcn_wmma_f32_16x16x4_f32(false, a, false, b, (short)0, acc3, false, false);
        }
    }

    // epilogue: optional ReLU, store 16x64 strip
    v8f accs[4] = {acc0, acc1, acc2, acc3};
#pragma unroll
    for (int j = 0; j < 4; ++j) {
        int col = j * 16 + l16;
#pragma unroll
        for (int r = 0; r < 8; ++r) {
            float v = accs[j][r];
            if (do_relu) v = fmaxf(v, 0.f);
            out[(size_t)(mtile * 16 + r + half * 8) * F_DIM + col] = v;
        }
    }
}

// ---------------- classifier: out = log_softmax(h @ wl + bl) ----------------
__global__ __launch_bounds__(256) void cls_kernel(
    const float* __restrict__ h, const float* __restrict__ wl,
    const float* __restrict__ bl, float* __restrict__ out, int n) {
    __shared__ float sw[F_DIM * CLS];
    __shared__ float sb[CLS];
    for (int i = threadIdx.x; i < F_DIM * CLS; i += blockDim.x) sw[i] = wl[i];
    if (threadIdx.x < CLS) sb[threadIdx.x] = bl[threadIdx.x];
    __syncthreads();

    int row = blockIdx.x * blockDim.x + threadIdx.x;
    if (row >= n) return;

    float acc[CLS];
#pragma unroll
    for (int c = 0; c < CLS; ++c) acc[c] = sb[c];

    const float4* hr = (const float4*)(h + (size_t)row * F_DIM);
#pragma unroll
    for (int k4 = 0; k4 < F_DIM / 4; ++k4) {
        float4 xv = hr[k4];
        const float* w = sw + k4 * 4 * CLS;
#pragma unroll
        for (int c = 0; c < CLS; ++c)
            acc[c] = fmaf(xv.x, w[c],
                     fmaf(xv.y, w[CLS + c],
                     fmaf(xv.z, w[2 * CLS + c],
                     fmaf(xv.w, w[3 * CLS + c], acc[c]))));
    }

    float mx = acc[0];
#pragma unroll
    for (int c = 1; c < CLS; ++c) mx = fmaxf(mx, acc[c]);
    float s = 0.f;
#pragma unroll
    for (int c = 0; c < CLS; ++c) s += __expf(acc[c] - mx);
    float lse = mx + __logf(s);
#pragma unroll
    for (int c = 0; c < CLS; ++c) out[(size_t)row * CLS + c] = acc[c] - lse;
}

// ---------------------------------------------------------------------------
extern "C" void kernel_launch(void* const* d_in, const int* in_sizes, int n_in,
                              void* d_out, int out_size, void* d_ws, size_t ws_size,
                              hipStream_t stream) {
    const float* x   = (const float*)d_in[0];
    const int*   ei  = (const int*)d_in[1];
    const float* w1  = (const float*)d_in[2];
    const float* b1  = (const float*)d_in[3];
    const float* w2  = (const float*)d_in[4];
    const float* b2  = (const float*)d_in[5];
    const float* wl  = (const float*)d_in[6];
    const float* bl  = (const float*)d_in[7];
    float* out = (float*)d_out;

    const int N = in_sizes[0] / F_DIM;     // 100000
    const int E = in_sizes[1] / 2;         // 1600000
    const int* src = ei;
    const int* dst = ei + E;

    // workspace carve-up (floats)
    float* norm = (float*)d_ws;                   // E
    float* dinv = norm + (size_t)E;               // N
    float* bufA = dinv + (size_t)N;               // N*64  (tx1 / h2)
    float* bufB = bufA + (size_t)N * F_DIM;       // N*64  (prop / tx2)
    float* bufC = bufB + (size_t)N * F_DIM;       // N*64  (hidden h1)

    const int  TPB    = 256;
    const long nd4    = (long)N * F_DIM / 4;
    const int  gz     = (int)((nd4 + TPB - 1) / TPB);
    const int  gN     = (N + TPB - 1) / TPB;
    const int  gE     = (E + TPB - 1) / TPB;
    const int  gSpmm  = (E * 32 + TPB - 1) / TPB; // one wave32 per edge
    const int  mtiles = N / 16;                   // N % 16 == 0
    const int  gGemm  = (mtiles + 7) / 8;         // 8 waves per block

    // ---- normalization: deg -> dinv -> norm --------------------------------
    zero_kernel<<<(N / 4 + TPB - 1) / TPB, TPB, 0, stream>>>((float4*)dinv, N / 4);
    degree_kernel<<<gE, TPB, 0, stream>>>(src, dst, dinv, E);
    dinv_kernel<<<gN, TPB, 0, stream>>>(dinv, N);
    norm_kernel<<<gE, TPB, 0, stream>>>(src, dst, dinv, norm, E);

    // ---- layer 1: tx0 = x --------------------------------------------------
    zero_kernel<<<gz, TPB, 0, stream>>>((float4*)bufA, nd4);
    spmm_kernel<<<gSpmm, TPB, 0, stream>>>(src, dst, norm, x, bufA, E);      // tx1
    zero_kernel<<<gz, TPB, 0, stream>>>((float4*)bufB, nd4);
    spmm_kernel<<<gSpmm, TPB, 0, stream>>>(src, dst, norm, bufA, bufB, E);
    combine_kernel<<<gz, TPB, 0, stream>>>((float4*)bufB, (const float4*)x, nd4); // tx2
    cheb_gemm_kernel<<<gGemm, TPB, 0, stream>>>(x, bufA, bufB, w1, b1, bufC,
                                                mtiles, /*relu=*/1);         // h1

    // ---- layer 2: tx0 = h1 -------------------------------------------------
    zero_kernel<<<gz, TPB, 0, stream>>>((float4*)bufA, nd4);
    spmm_kernel<<<gSpmm, TPB, 0, stream>>>(src, dst, norm, bufC, bufA, E);   // tx1
    zero_kernel<<<gz, TPB, 0, stream>>>((float4*)bufB, nd4);
    spmm_kernel<<<gSpmm, TPB, 0, stream>>>(src, dst, norm, bufA, bufB, E);
    combine_kernel<<<gz, TPB, 0, stream>>>((float4*)bufB, (const float4*)bufC, nd4); // tx2
    // write h2 over bufA: each wave reads its own 16-row strip of bufA fully
    // before storing the same rows -> race-free in-place use.
    cheb_gemm_kernel<<<gGemm, TPB, 0, stream>>>(bufC, bufA, bufB, w2, b2, bufA,
                                                mtiles, /*relu=*/0);         // h2

    // ---- classifier + log_softmax -----------------------------------------
    cls_kernel<<<gN, TPB, 0, stream>>>(bufA, wl, bl, out, N);

    // ---- second tuple element: pass-through edge_index (bit copy) ----------
    hipMemcpyAsync((char*)d_out + (size_t)N * CLS * sizeof(float), ei,
                   (size_t)2 * E * sizeof(int), hipMemcpyDeviceToDevice, stream);
}